// LQRSolver_7456063226388
// MI455X (gfx1250) — compile-verified
//
#include <hip/hip_runtime.h>

// ---------------------------------------------------------------------------
// LQR backward Riccati recursion on MI455X (gfx1250, wave32).
// Persistent single-workgroup kernel: T=2048 serial steps, all fp32 GEMM work
// expressed as V_WMMA_F32_16X16X4_F32 tiles. 16 waves (512 threads).
// ---------------------------------------------------------------------------

typedef __attribute__((ext_vector_type(2))) float v2f;
typedef __attribute__((ext_vector_type(8))) float v8f;

#define NS   128            // n (state)
#define NC   64             // m (control)
#define DT   192            // d = n + m

// ---- LDS layout (float offsets), total 70,656 floats = 282,624 bytes ------
#define O_F    0                       // F : 128 x 192 (persistent)
#define O_FV   24576                   // FV = F^T V : 192 x 128
#define O_U    49152                   // union region (20480 floats):
                                       //  phase A: V 128x128 (16384)
#define O_QUX  (O_U)                   //  phase B: Qux 64x128 (8192)
#define O_QUU  (O_U + 8192)            //           Quu 64x64  (4096)
#define O_K    (O_U + 12288)           //           K   64x128 (8192)
#define O_MISC (O_U + 20480)           // vectors / scalars
#define O_qv   (O_MISC)                // q    : 192
#define O_fv   (O_MISC + 192)          // f    : 128
#define O_cv   (O_MISC + 320)          // c    : 192
#define O_vold (O_MISC + 512)          // v    : 128
#define O_kv   (O_MISC + 640)          // k    : 64
#define O_wt   (O_MISC + 704)          // V@f  : 128
#define O_sc   (O_MISC + 832)          // [0]=const accum, [1]=partial c_step
#define LDS_FLOATS (O_MISC + 1024)     // 70656
#define LDS_BYTES  (LDS_FLOATS * 4)    // 282624 (< 320K WGP LDS)

// ---- fp32 WMMA 16x16x4 helpers --------------------------------------------
// A frag (16x4): lane l (row = l%16): V0 = A[row][k0+2*(l/16)], V1 = +1
// B frag (4x16): lane l (col = l%16): V0 = B[k0+2*(l/16)][col], V1 = +1
// C/D frag      : elem e -> (row = 8*(l/16)+e, col = l%16)
__device__ __forceinline__ v2f frag_a(const float* m, int ld, int r0, int k0,
                                      int row, int kk) {
  v2f a;
  a[0] = m[(r0 + row) * ld + k0 + kk];
  a[1] = m[(r0 + row) * ld + k0 + kk + 1];
  return a;
}
// A^T fragment: logical A[r][k] = m[k][r]  (used for F^T and Qux^T)
__device__ __forceinline__ v2f frag_at(const float* m, int ld, int r0, int k0,
                                       int row, int kk) {
  v2f a;
  a[0] = m[(k0 + kk) * ld + r0 + row];
  a[1] = m[(k0 + kk + 1) * ld + r0 + row];
  return a;
}
__device__ __forceinline__ v2f frag_b(const float* m, int ld, int k0, int c0,
                                      int col, int kk) {
  v2f b;
  b[0] = m[(k0 + kk) * ld + c0 + col];
  b[1] = m[(k0 + kk + 1) * ld + c0 + col];
  return b;
}
__device__ __forceinline__ v8f wmma4(v2f a, v2f b, v8f c) {
  // (neg_a, A, neg_b, B, c_mod, C, reuse_a, reuse_b)
  return __builtin_amdgcn_wmma_f32_16x16x4_f32(false, a, false, b, (short)0, c,
                                               false, false);
}

__global__ __launch_bounds__(512, 1) void lqr_riccati_kernel(
    const float* __restrict__ F,   // 128 x 192
    const float* __restrict__ fv,  // 128
    const float* __restrict__ Cm,  // 192 x 192
    const float* __restrict__ cv,  // 192
    const int* __restrict__ Tp,
    float* __restrict__ out) {
  extern __shared__ float lds[];

  const int tid  = threadIdx.x;
  const int lane = tid & 31;
  const int wave = tid >> 5;
  const int NW   = blockDim.x >> 5;  // 16 waves
  const int T    = *Tp;

  const int row  = lane & 15;        // A row / B,C,D col within tile
  const int half = lane >> 4;
  const int kk   = 2 * half;         // K sub-offset for A/B frags
  const int crow = 8 * half;         // base row of C/D frag

  // output stack offsets (flat concat: Ks, ks, Vs, vs, consts)
  const size_t oKs = 0;
  const size_t oks = (size_t)T * (NC * NS);
  const size_t oVs = oks + (size_t)T * NC;
  const size_t ovs = oVs + (size_t)T * (NS * NS);
  const size_t ocs = ovs + (size_t)T * NS;

  // ---- one-time staging: F, f, c into LDS; v0 = c[:128] --------------------
  for (int i = tid; i < NS * DT; i += blockDim.x) lds[O_F + i] = F[i];
  for (int i = tid; i < NS; i += blockDim.x) {
    lds[O_fv + i]   = fv[i];
    lds[O_vold + i] = cv[i];
  }
  for (int i = tid; i < DT; i += blockDim.x) lds[O_cv + i] = cv[i];
  if (tid == 0) lds[O_sc] = 0.0f;
  __syncthreads();

  for (int t = 0; t < T; ++t) {
    const int r = T - 1 - t;  // time-reversed output index

    // ---- Stage 0: stage V into LDS union (phase A). t==0: V0 = C[:128,:128]
    const float* vsrc;
    int vld;
    if (t == 0) { vsrc = Cm; vld = DT; }
    else        { vsrc = out + oVs + (size_t)(r + 1) * (NS * NS); vld = NS; }
    for (int i = tid; i < NS * NS; i += blockDim.x) {
      int rr = i >> 7, cc = i & 127;
      lds[O_U + i] = vsrc[(size_t)rr * vld + cc];
    }
    __syncthreads();

    // ---- Stage 1: FV = F^T @ V  (192x128): 12x8 tiles, K=128 --------------
    for (int tile = wave; tile < 96; tile += NW) {
      int ti = tile >> 3, tj = tile & 7;
      v8f acc = {};
      for (int kb = 0; kb < 32; ++kb) {
        v2f a = frag_at(lds + O_F, DT, ti * 16, kb * 4, row, kk);  // F^T
        v2f b = frag_b(lds + O_U, NS, kb * 4, tj * 16, row, kk);   // V
        acc = wmma4(a, b, acc);
      }
#pragma unroll
      for (int e = 0; e < 8; ++e)
        lds[O_FV + (ti * 16 + crow + e) * NS + tj * 16 + row] = acc[e];
    }
    __syncthreads();

    // ---- Stage 2: q = c + FV@f + F^T@v ;  wt = V@f  (VALU) ----------------
    if (tid < DT) {
      float s = lds[O_cv + tid];
      for (int j = 0; j < NS; ++j) {
        s += lds[O_FV + tid * NS + j] * lds[O_fv + j];
        s += lds[O_F + j * DT + tid] * lds[O_vold + j];
      }
      lds[O_qv + tid] = s;
    } else if (tid >= 256 && tid < 256 + NS) {
      int i = tid - 256;
      float s = 0.f;
      for (int j = 0; j < NS; ++j) s += lds[O_U + i * NS + j] * lds[O_fv + j];
      lds[O_wt + i] = s;
    }
    __syncthreads();
    if (tid == 0) {  // partial c_step = 0.5 f^T V f + f^T v
      float s1 = 0.f, s2 = 0.f;
      for (int i = 0; i < NS; ++i) {
        s1 += lds[O_fv + i] * lds[O_wt + i];
        s2 += lds[O_fv + i] * lds[O_vold + i];
      }
      lds[O_sc + 1] = 0.5f * s1 + s2;
    }
    __syncthreads();  // V (phase A) dead; union switches to phase B

    // ---- Stage 3: Qux (4x8 tiles), Quu (4x4 tiles) = C + FV@F (rows 128..)
    for (int tile = wave; tile < 48; tile += NW) {
      int lr0, lc0, gc0, dstO, dstLd;
      if (tile < 32) {
        int ti = tile >> 3, tj = tile & 7;
        lr0 = ti * 16; lc0 = tj * 16; gc0 = lc0; dstO = O_QUX; dstLd = NS;
      } else {
        int t2 = tile - 32, ti = t2 >> 2, tj = t2 & 3;
        lr0 = ti * 16; lc0 = tj * 16; gc0 = NS + lc0; dstO = O_QUU; dstLd = NC;
      }
      int gr0 = NS + lr0;
      v8f acc;
#pragma unroll
      for (int e = 0; e < 8; ++e)
        acc[e] = Cm[(size_t)(gr0 + crow + e) * DT + gc0 + row];
      for (int kb = 0; kb < 32; ++kb) {
        v2f a = frag_a(lds + O_FV, NS, gr0, kb * 4, row, kk);  // FV (u rows)
        v2f b = frag_b(lds + O_F, DT, kb * 4, gc0, row, kk);   // F cols
        acc = wmma4(a, b, acc);
      }
#pragma unroll
      for (int e = 0; e < 8; ++e)
        lds[dstO + (lr0 + crow + e) * dstLd + lc0 + row] = acc[e];
    }
    __syncthreads();

    // ---- Stage 4: Quu <- Quu^{-1}, in-place Gauss-Jordan (SPD, no pivot) --
    // 4096 elements, 8 per thread, 2 barriers per pivot.
    for (int p = 0; p < NC; ++p) {
      float invd = 1.0f / lds[O_QUU + p * NC + p];
      float nv[8];
#pragma unroll
      for (int e = 0; e < 8; ++e) {
        int idx = tid * 8 + e, i = idx >> 6, j = idx & 63;
        float a_ij = lds[O_QUU + idx];
        float a_ip = lds[O_QUU + i * NC + p];
        float a_pj = lds[O_QUU + p * NC + j];
        if (i == p && j == p)      nv[e] = invd;
        else if (i == p)           nv[e] = a_pj * invd;
        else if (j == p)           nv[e] = -a_ip * invd;
        else                       nv[e] = a_ij - a_ip * invd * a_pj;
      }
      __syncthreads();
#pragma unroll
      for (int e = 0; e < 8; ++e) lds[O_QUU + tid * 8 + e] = nv[e];
      __syncthreads();
    }

    // ---- Stage 5: K = -Quu^{-1} @ Qux (4x8 tiles, K=64); k = -Quu^{-1} qu -
    float* Kout = out + oKs + (size_t)r * (NC * NS);
    for (int tile = wave; tile < 32; tile += NW) {
      int ti = tile >> 3, tj = tile & 7;
      v8f acc = {};
      for (int kb = 0; kb < 16; ++kb) {
        v2f a = frag_a(lds + O_QUU, NC, ti * 16, kb * 4, row, kk);
        v2f b = frag_b(lds + O_QUX, NS, kb * 4, tj * 16, row, kk);
        acc = wmma4(a, b, acc);
      }
#pragma unroll
      for (int e = 0; e < 8; ++e) {
        float v = -acc[e];
        lds[O_K + (ti * 16 + crow + e) * NS + tj * 16 + row] = v;
        Kout[(ti * 16 + crow + e) * NS + tj * 16 + row] = v;
      }
    }
    __syncthreads();
    if (tid < NC) {
      float s = 0.f;
      for (int j = 0; j < NC; ++j)
        s -= lds[O_QUU + tid * NC + j] * lds[O_qv + NS + j];
      lds[O_kv + tid] = s;
      out[oks + (size_t)r * NC + tid] = s;
    }
    __syncthreads();

    // ---- Stage 5b: c_step = 0.5 k^T qu + (0.5 f^T V f + f^T v); accumulate
    if (tid == 0) {
      float s = 0.f;
      for (int u = 0; u < NC; ++u) s += lds[O_kv + u] * lds[O_qv + NS + u];
      float ca = lds[O_sc] + 0.5f * s + lds[O_sc + 1];
      lds[O_sc] = ca;
      out[ocs + r] = ca;
    }
    __syncthreads();

    // ---- Stage 6: Vn = Qxx + Qxu@K, Qxx fused = Cxx + FV[0:128]@F[:,0:128]
    // (uses Quu K = -Qux  =>  Qxx + QxuK + K^TQux + K^TQuuK == Qxx + QxuK)
    float* Vout = out + oVs + (size_t)r * (NS * NS);
    for (int tile = wave; tile < 64; tile += NW) {
      int ti = tile >> 3, tj = tile & 7;
      v8f acc;
#pragma unroll
      for (int e = 0; e < 8; ++e)
        acc[e] = Cm[(size_t)(ti * 16 + crow + e) * DT + tj * 16 + row];
      for (int kb = 0; kb < 32; ++kb) {  // Qxx part, K=128
        v2f a = frag_a(lds + O_FV, NS, ti * 16, kb * 4, row, kk);
        v2f b = frag_b(lds + O_F, DT, kb * 4, tj * 16, row, kk);
        acc = wmma4(a, b, acc);
      }
      for (int kb = 0; kb < 16; ++kb) {  // Qxu@K part, K=64 (Qxu = Qux^T)
        v2f a = frag_at(lds + O_QUX, NS, ti * 16, kb * 4, row, kk);
        v2f b = frag_b(lds + O_K, NS, kb * 4, tj * 16, row, kk);
        acc = wmma4(a, b, acc);
      }
#pragma unroll
      for (int e = 0; e < 8; ++e)
        Vout[(ti * 16 + crow + e) * NS + tj * 16 + row] = acc[e];
    }
    __syncthreads();

    // ---- Stage 7: vn = qx + Qxu@k; becomes v for next step ----------------
    if (tid < NS) {
      float s = lds[O_qv + tid];
      for (int u = 0; u < NC; ++u)
        s += lds[O_QUX + u * NS + tid] * lds[O_kv + u];
      out[ovs + (size_t)r * NS + tid] = s;
      lds[O_vold + tid] = s;
    }
    __threadfence();   // make Vout visible before next-step stage 0 reads it
    __syncthreads();
  }
}

extern "C" void kernel_launch(void* const* d_in, const int* in_sizes, int n_in,
                              void* d_out, int out_size, void* d_ws,
                              size_t ws_size, hipStream_t stream) {
  (void)in_sizes; (void)n_in; (void)d_ws; (void)ws_size; (void)out_size;
  const float* F  = (const float*)d_in[0];
  const float* f  = (const float*)d_in[1];
  const float* C  = (const float*)d_in[2];
  const float* c  = (const float*)d_in[3];
  const int*   T  = (const int*)d_in[4];
  float* out = (float*)d_out;

  static bool attr_set = false;  // idempotent host-side attribute, not a stream op
  if (!attr_set) {
    hipFuncSetAttribute((const void*)lqr_riccati_kernel,
                        hipFuncAttributeMaxDynamicSharedMemorySize, LDS_BYTES);
    attr_set = true;
  }
  lqr_riccati_kernel<<<1, 512, LDS_BYTES, stream>>>(F, f, C, c, T, out);
}